// ModulatedDeformConv_13615046329034
// MI455X (gfx1250) — compile-verified
//
#include <hip/hip_runtime.h>

// Modulated deformable conv (DCNv2) fused im2col + WMMA GEMM for gfx1250.
// B=8, C=64, H=W=128, OC=64, K=3x3, stride=1, pad=1, dil=1, groups=1.
//
// v2: 64-position tile per workgroup, 2 N-subtiles per wave (A-fragment reuse
// across two v_wmma_f32_16x16x4_f32), position-major LDS with 580-word row
// stride -> single conflict-free ds_load_b64 per B fragment in the hot loop.

typedef __attribute__((ext_vector_type(2))) float v2f;
typedef __attribute__((ext_vector_type(8))) float v8f;

namespace {
constexpr int Bn  = 8;
constexpr int Cn  = 64;
constexpr int Hn  = 128;
constexpr int Wn  = 128;
constexpr int OCn = 64;
constexpr int Kn  = 9;                 // 3x3 taps
constexpr int HOn = 128;
constexpr int WOn = 128;
constexpr int CK  = Cn * Kn;           // 576 reduction depth
constexpr int TILE_N = 64;             // output positions per workgroup
constexpr int LDS_STRIDE = 580;        // CK + 4 pad; %64==4 -> conflict-free b64 loads
constexpr int HW  = Hn * Wn;           // 16384
constexpr int NPB = HOn * WOn;         // 16384 positions per (b, oc)
}

__global__ __launch_bounds__(256)
void mdconv_wmma_f32_kernel(const float* __restrict__ in,     // [B,C,H,W]
                            const float* __restrict__ off,    // [B,2K,HO,WO]
                            const float* __restrict__ msk,    // [B,K,HO,WO]
                            const float* __restrict__ wgt,    // [OC,C,3,3] = [OC,CK]
                            const float* __restrict__ bias,   // [OC]
                            float* __restrict__ out)          // [B,OC,HO,WO]
{
    extern __shared__ float smem[];          // cols[TILE_N][LDS_STRIDE], pos-major

    const int lane = threadIdx.x & 31;
    const int wave = threadIdx.x >> 5;       // 0..7
    const int blk  = blockIdx.x;
    const int b      = blk >> 8;             // 256 tiles per batch image
    const int n_base = (blk & 255) * TILE_N; // 64 positions, same output row
    const int p      = n_base >> 7;          // output row (64 | 128)
    const int q_base = n_base & 127;         // 0 or 64

    // ---------------- Phase 1: im2col tile into LDS (position-major) --------
    // waves 0-3: positions 0..31, channels (wave&3)*16..+16
    // waves 4-7: positions 32..63, same channel split
    {
        const int j  = ((wave >> 2) << 5) + lane;   // position in tile, 0..63
        const int q  = q_base + j;
        const int c0 = (wave & 3) * 16;
        const float* inB = in + (size_t)b * Cn * HW;

        #pragma unroll
        for (int k = 0; k < Kn; ++k) {
            const int kyi = k / 3, kxi = k % 3;
            const float offy = off[((size_t)(b * 18 + 2 * k)     * HOn + p) * WOn + q];
            const float offx = off[((size_t)(b * 18 + 2 * k + 1) * HOn + p) * WOn + q];
            const float m    = msk[((size_t)(b * 9  + k)         * HOn + p) * WOn + q];

            const float py = (float)(p - 1 + kyi) + offy;
            const float px = (float)(q - 1 + kxi) + offx;
            const float fy = floorf(py), fx = floorf(px);
            const int   y0 = (int)fy,    x0 = (int)fx;
            const float wy1 = py - fy,   wx1 = px - fx;
            const float wy0 = 1.0f - wy1, wx0 = 1.0f - wx1;

            const int y1 = y0 + 1, x1 = x0 + 1;
            const float vy0 = (y0 >= 0 && y0 < Hn) ? 1.0f : 0.0f;
            const float vy1 = (y1 >= 0 && y1 < Hn) ? 1.0f : 0.0f;
            const float vx0 = (x0 >= 0 && x0 < Wn) ? 1.0f : 0.0f;
            const float vx1 = (x1 >= 0 && x1 < Wn) ? 1.0f : 0.0f;

            const int y0c = min(max(y0, 0), Hn - 1);
            const int y1c = min(max(y1, 0), Hn - 1);
            const int x0c = min(max(x0, 0), Wn - 1);
            const int x1c = min(max(x1, 0), Wn - 1);

            // fold mask + zero-padding validity into corner weights
            const float w00 = wy0 * wx0 * m * vy0 * vx0;
            const float w01 = wy0 * wx1 * m * vy0 * vx1;
            const float w10 = wy1 * wx0 * m * vy1 * vx0;
            const float w11 = wy1 * wx1 * m * vy1 * vx1;

            const int i00 = y0c * Wn + x0c;
            const int i01 = y0c * Wn + x1c;
            const int i10 = y1c * Wn + x0c;
            const int i11 = y1c * Wn + x1c;

            #pragma unroll
            for (int cc = 0; cc < 16; ++cc) {
                const int c = c0 + cc;
                const float* img = inB + (size_t)c * HW;
                const float v = img[i00] * w00 + img[i01] * w01 +
                                img[i10] * w10 + img[i11] * w11;
                smem[j * LDS_STRIDE + c * Kn + k] = v;   // cols[pos=j][K=c*9+k]
            }
        }
    }
    __syncthreads();

    // ---------------- Phase 2: per-wave 16x32 output via 2 WMMA tiles -------
    // wave&3 -> M tile; wave>>2 -> N half (32 positions); 2 accumulators.
    const int m0   = (wave & 3) * 16;
    const int nh   = wave >> 2;             // 0 or 1
    const int half = lane >> 4;             // ISA half: lanes 0-15 / 16-31
    const int l15  = lane & 15;
    const int kOff = half * 2;              // halves hold K {0,1} / {2,3}

    const float* aptr = wgt + (size_t)(m0 + l15) * CK + kOff;   // 8B aligned
    const float* b0p  = smem + (nh * 32 + l15) * LDS_STRIDE + kOff;
    const float* b1p  = b0p + 16 * LDS_STRIDE;

    v8f acc0 = {0.f, 0.f, 0.f, 0.f, 0.f, 0.f, 0.f, 0.f};
    v8f acc1 = {0.f, 0.f, 0.f, 0.f, 0.f, 0.f, 0.f, 0.f};

    #pragma unroll 4
    for (int kk = 0; kk < CK; kk += 4) {
        const v2f a  = *reinterpret_cast<const v2f*>(aptr + kk);  // global, L2-resident
        const v2f b0 = *reinterpret_cast<const v2f*>(b0p + kk);   // ds_load_b64
        const v2f b1 = *reinterpret_cast<const v2f*>(b1p + kk);   // ds_load_b64
        acc0 = __builtin_amdgcn_wmma_f32_16x16x4_f32(false, a, false, b0,
                                                     (short)0, acc0, false, false);
        acc1 = __builtin_amdgcn_wmma_f32_16x16x4_f32(false, a, false, b1,
                                                     (short)0, acc1, false, false);
    }

    // ---------------- Store: ISA C/D layout -> out[b,oc,p,q] ----------------
    float* outB = out + (size_t)b * OCn * NPB;
    const int n0 = n_base + nh * 32 + l15;
    #pragma unroll
    for (int v = 0; v < 8; ++v) {
        const int oc = m0 + half * 8 + v;    // VGPR v, lane half -> row M
        const float bv = bias[oc];
        outB[(size_t)oc * NPB + n0]      = acc0[v] + bv;
        outB[(size_t)oc * NPB + n0 + 16] = acc1[v] + bv;
    }
}

extern "C" void kernel_launch(void* const* d_in, const int* in_sizes, int n_in,
                              void* d_out, int out_size, void* d_ws, size_t ws_size,
                              hipStream_t stream) {
    (void)in_sizes; (void)n_in; (void)out_size; (void)d_ws; (void)ws_size;
    const float* in   = (const float*)d_in[0];
    const float* off  = (const float*)d_in[1];
    const float* msk  = (const float*)d_in[2];
    const float* wgt  = (const float*)d_in[3];
    const float* bias = (const float*)d_in[4];
    float* out = (float*)d_out;

    const int blocks = Bn * (NPB / TILE_N);                       // 8 * 256 = 2048
    const size_t lds = (size_t)TILE_N * LDS_STRIDE * sizeof(float); // 148480 B
    mdconv_wmma_f32_kernel<<<blocks, 256, lds, stream>>>(in, off, msk, wgt, bias, out);
}